// GAT_79491254714995
// MI455X (gfx1250) — compile-verified
//
#include <hip/hip_runtime.h>
#include <hip/hip_bf16.h>

// ---------------- problem constants (match reference) ----------------
#define NNODE 10000
#define NEDGE 160000
#define ETOT  (NEDGE + NNODE)   // edges + self loops
#define FIN   256
#define HIDC  128
#define H1H   5
#define H3H   3
#define NCLS  10
#define NG    64
#define NEG_SLOPE 0.2f

typedef __attribute__((ext_vector_type(2))) float v2f;
typedef __attribute__((ext_vector_type(8))) float v8f;

// ---------------- utility ----------------
__global__ void fill_kernel(float* __restrict__ p, float v, unsigned n) {
    unsigned i = blockIdx.x * blockDim.x + threadIdx.x;
    if (i < n) p[i] = v;
}

__device__ __forceinline__ float atomicMaxFloat(float* addr, float value) {
    // ordering trick: works for mixed-sign floats given -inf-ish init
    if (value >= 0.0f)
        return __int_as_float(atomicMax((int*)addr, __float_as_int(value)));
    else
        return __uint_as_float(atomicMin((unsigned int*)addr, __float_as_uint(value)));
}

__device__ __forceinline__ float leaky(float x) {
    return x > 0.0f ? x : NEG_SLOPE * x;
}

// ---------------- fp32 WMMA GEMM: C[M,NC] = A[M,K] @ B[K,NC] ----------------
// K, NC compile-time -> B strides become immediate offsets, no address math
// in the hot loop. One wave computes a 16(M) x 64(N) tile; K stepped by 4
// with v_wmma_f32_16x16x4_f32 (full fp32 precision, matches reference).
template <int K, int NC>
__global__ void gemm_wmma_f32_kernel(const float* __restrict__ A,
                                     const float* __restrict__ B,
                                     float* __restrict__ C, int M) {
    const int lane = threadIdx.x & 31;
    const int r    = lane & 15;           // A row / B col within tile
    const int kk   = (lane >> 4) * 2;     // K sub-offset per ISA 16x4 layout
    const int m0   = (blockIdx.x * 4 + threadIdx.y) * 16;
    const int n0   = blockIdx.y * 64;

    const int ar = min(m0 + r, M - 1);    // clamp reads; guard writes below
    const float* Ap = A + (size_t)ar * K + kk;
    const float* Bp = B + (size_t)kk * NC + n0 + r;

    v8f acc0 = {}, acc1 = {}, acc2 = {}, acc3 = {};

#pragma unroll 2
    for (int k = 0; k < K; k += 4) {
        if (k + 16 < K) {
            // pull the next B panel into the WGP-level cache
            __builtin_prefetch(Bp + 16 * NC, 0, 3);
        }
        v2f a;
        a.x = Ap[0];
        a.y = Ap[1];

        v2f b;
        b.x = Bp[0];       b.y = Bp[NC];
        acc0 = __builtin_amdgcn_wmma_f32_16x16x4_f32(false, a, false, b, (short)0, acc0, false, false);
        b.x = Bp[16];      b.y = Bp[NC + 16];
        acc1 = __builtin_amdgcn_wmma_f32_16x16x4_f32(false, a, false, b, (short)0, acc1, false, false);
        b.x = Bp[32];      b.y = Bp[NC + 32];
        acc2 = __builtin_amdgcn_wmma_f32_16x16x4_f32(false, a, false, b, (short)0, acc2, false, false);
        b.x = Bp[48];      b.y = Bp[NC + 48];
        acc3 = __builtin_amdgcn_wmma_f32_16x16x4_f32(false, a, false, b, (short)0, acc3, false, false);

        Ap += 4;
        Bp += 4 * NC;
    }

    // D layout: VGPR j -> M = j (lanes 0-15), M = j+8 (lanes 16-31); N = lane&15
    const int rowBase = m0 + ((lane >> 4) * 8);
    const int col     = n0 + (lane & 15);
#pragma unroll
    for (int j = 0; j < 8; ++j) {
        int row = rowBase + j;
        if (row < M) {
            C[(size_t)row * NC + col +  0] = acc0[j];
            C[(size_t)row * NC + col + 16] = acc1[j];
            C[(size_t)row * NC + col + 32] = acc2[j];
            C[(size_t)row * NC + col + 48] = acc3[j];
        }
    }
}

// ---------------- attention logit dot products ----------------
// one wave per (node, head): al_s[n,h] = <h[n,h,:], a_src[h,:]>, same for a_dst
// float4 loads: one global_load_b128 per lane covers the 128-wide feature row
__global__ void attn_dots_kernel(const float* __restrict__ hbuf,
                                 const float* __restrict__ a_s,
                                 const float* __restrict__ a_d,
                                 float* __restrict__ als,
                                 float* __restrict__ ald, int H) {
    unsigned t = blockIdx.x * blockDim.x + threadIdx.x;
    unsigned wave = t >> 5;
    int lane = t & 31;
    unsigned total = (unsigned)NNODE * H;
    if (wave >= total) return;
    int n  = wave / H;
    int hh = wave - n * H;

    const float4 hv = *(const float4*)(hbuf + ((size_t)n * H + hh) * HIDC + lane * 4);
    const float4 as = *(const float4*)(a_s + hh * HIDC + lane * 4);
    const float4 ad = *(const float4*)(a_d + hh * HIDC + lane * 4);

    float s1 = hv.x * as.x + hv.y * as.y + hv.z * as.z + hv.w * as.w;
    float s2 = hv.x * ad.x + hv.y * ad.y + hv.z * ad.z + hv.w * ad.w;
#pragma unroll
    for (int off = 16; off > 0; off >>= 1) {
        s1 += __shfl_down(s1, off, 32);
        s2 += __shfl_down(s2, off, 32);
    }
    if (lane == 0) {
        als[wave] = s1;
        ald[wave] = s2;
    }
}

// ---------------- segment softmax pass 1: max over dst ----------------
__global__ void edge_max_kernel(const int* __restrict__ ei,
                                const float* __restrict__ als,
                                const float* __restrict__ ald,
                                float* __restrict__ mbuf, int H) {
    unsigned t = blockIdx.x * blockDim.x + threadIdx.x;
    unsigned total = (unsigned)ETOT * H;
    if (t >= total) return;
    int e  = t / H;
    int hh = t - e * H;
    int s = (e < NEDGE) ? ei[e]         : (e - NEDGE);
    int d = (e < NEDGE) ? ei[NEDGE + e] : (e - NEDGE);
    float lg = leaky(als[s * H + hh] + ald[d * H + hh]);
    atomicMaxFloat(&mbuf[d * H + hh], lg);
}

// ---------------- pass 2: exp + segment sum ----------------
__global__ void edge_exp_kernel(const int* __restrict__ ei,
                                const float* __restrict__ als,
                                const float* __restrict__ ald,
                                const float* __restrict__ mbuf,
                                float* __restrict__ sbuf,
                                float* __restrict__ exb, int H) {
    unsigned t = blockIdx.x * blockDim.x + threadIdx.x;
    unsigned total = (unsigned)ETOT * H;
    if (t >= total) return;
    int e  = t / H;
    int hh = t - e * H;
    int s = (e < NEDGE) ? ei[e]         : (e - NEDGE);
    int d = (e < NEDGE) ? ei[NEDGE + e] : (e - NEDGE);
    float lg = leaky(als[s * H + hh] + ald[d * H + hh]);
    float ex = __expf(lg - mbuf[d * H + hh]);
    exb[t] = ex;
    atomicAdd(&sbuf[d * H + hh], ex);
}

// ---------------- pass 3: normalized weighted aggregation ----------------
// one wave per (edge, head); float4 gather + 4 coalesced atomic f32 adds/lane
__global__ void edge_agg_kernel(const int* __restrict__ ei,
                                const float* __restrict__ exb,
                                const float* __restrict__ sbuf,
                                const float* __restrict__ hbuf,
                                float* __restrict__ agg, int H) {
    unsigned long long t = (unsigned long long)blockIdx.x * blockDim.x + threadIdx.x;
    unsigned wave = (unsigned)(t >> 5);
    int lane = (int)(t & 31);
    unsigned total = (unsigned)ETOT * H;
    if (wave >= total) return;
    int e  = wave / H;
    int hh = wave - e * H;
    int s = (e < NEDGE) ? ei[e]         : (e - NEDGE);
    int d = (e < NEDGE) ? ei[NEDGE + e] : (e - NEDGE);
    float w = exb[wave] / (sbuf[d * H + hh] + 1e-16f);
    const float4 hv = *(const float4*)(hbuf + ((size_t)s * H + hh) * HIDC + lane * 4);
    float* ag = agg + ((size_t)d * H + hh) * HIDC + lane * 4;
    atomicAdd(&ag[0], w * hv.x);
    atomicAdd(&ag[1], w * hv.y);
    atomicAdd(&ag[2], w * hv.z);
    atomicAdd(&ag[3], w * hv.w);
}

// ---------------- layer finalizers ----------------
__global__ void fin1_kernel(const float* __restrict__ agg,
                            const float* __restrict__ b,
                            float* __restrict__ x1, unsigned n) {
    unsigned i = blockIdx.x * blockDim.x + threadIdx.x;
    if (i >= n) return;
    float v = agg[i] + b[i % (H1H * HIDC)];
    x1[i] = v > 0.0f ? v : __expf(v) - 1.0f;   // elu
}

__global__ void fin2_kernel(const float* __restrict__ x1,
                            const float* __restrict__ agg,
                            const float* __restrict__ b,
                            float* __restrict__ x2, unsigned n) {
    unsigned i = blockIdx.x * blockDim.x + threadIdx.x;
    if (i >= n) return;
    float v = x1[i] + agg[i] + b[i % (H1H * HIDC)];
    x2[i] = v > 0.0f ? v : __expf(v) - 1.0f;   // elu(x1 + h)
}

__global__ void fin3_kernel(const float* __restrict__ agg,
                            const float* __restrict__ b,
                            float* __restrict__ x3, unsigned n) {
    unsigned i = blockIdx.x * blockDim.x + threadIdx.x;
    if (i >= n) return;
    int node = i / HIDC;
    int c    = i - node * HIDC;
    float s = 0.0f;
#pragma unroll
    for (int hh = 0; hh < H3H; ++hh)
        s += agg[((size_t)node * H3H + hh) * HIDC + c];
    x3[i] = s * (1.0f / H3H) + b[c];           // head mean + bias
}

// ---------------- global mean pool ----------------
__global__ void pool_kernel(const float* __restrict__ x3,
                            const int* __restrict__ batch,
                            float* __restrict__ pooled,
                            float* __restrict__ counts) {
    unsigned i = blockIdx.x * blockDim.x + threadIdx.x;
    unsigned n = (unsigned)NNODE * HIDC;
    if (i >= n) return;
    int node = i / HIDC;
    int c    = i - node * HIDC;
    int g    = batch[node];
    atomicAdd(&pooled[g * HIDC + c], x3[i]);
    if (c == 0) atomicAdd(&counts[g], 1.0f);
}

// ---------------- classifier + log_softmax (single block) ----------------
__global__ void classify_kernel(const float* __restrict__ pooled,
                                const float* __restrict__ counts,
                                const float* __restrict__ Wc,
                                const float* __restrict__ bc,
                                float* __restrict__ out) {
    __shared__ float sl[NG][NCLS];
    __shared__ float srow[NG][2];
    int tx = threadIdx.x;   // class (0..15, active < NCLS)
    int g  = threadIdx.y;   // graph
    if (tx < NCLS) {
        float inv = 1.0f / fmaxf(counts[g], 1.0f);
        float acc = bc[tx];
        for (int k = 0; k < HIDC; ++k)
            acc += pooled[g * HIDC + k] * inv * Wc[k * NCLS + tx];
        sl[g][tx] = acc;
    }
    __syncthreads();
    if (tx == 0) {
        float mx = -3.0e38f;
        for (int c = 0; c < NCLS; ++c) mx = fmaxf(mx, sl[g][c]);
        float s = 0.0f;
        for (int c = 0; c < NCLS; ++c) s += __expf(sl[g][c] - mx);
        srow[g][0] = mx;
        srow[g][1] = logf(s);
    }
    __syncthreads();
    if (tx < NCLS) {
        float lg = sl[g][tx];
        out[g * NCLS + tx] = lg;                              // logits
        out[NG * NCLS + g * NCLS + tx] = lg - srow[g][0] - srow[g][1]; // log_softmax
    }
}

// ---------------- host helpers ----------------
static inline unsigned cdiv(unsigned a, unsigned b) { return (a + b - 1) / b; }

// attention + segment-softmax + aggregation (GEMM is launched by caller,
// since its template instantiation depends on layer shapes)
static void run_attention(int Hh, const float* a_s, const float* a_d, const int* ei,
                          float* hbuf, float* als, float* ald, float* mbuf,
                          float* sbuf, float* exb, float* agg, hipStream_t stream) {
    const int Nout = Hh * HIDC;

    unsigned nhw = (unsigned)NNODE * Hh;                       // waves
    attn_dots_kernel<<<cdiv(nhw * 32, 256), 256, 0, stream>>>(hbuf, a_s, a_d, als, ald, Hh);

    fill_kernel<<<cdiv((unsigned)NNODE * Hh, 256), 256, 0, stream>>>(mbuf, -3.0e38f, (unsigned)NNODE * Hh);
    fill_kernel<<<cdiv((unsigned)NNODE * Hh, 256), 256, 0, stream>>>(sbuf, 0.0f, (unsigned)NNODE * Hh);
    fill_kernel<<<cdiv((unsigned)NNODE * Nout, 256), 256, 0, stream>>>(agg, 0.0f, (unsigned)NNODE * Nout);

    unsigned ehn = (unsigned)ETOT * Hh;
    edge_max_kernel<<<cdiv(ehn, 256), 256, 0, stream>>>(ei, als, ald, mbuf, Hh);
    edge_exp_kernel<<<cdiv(ehn, 256), 256, 0, stream>>>(ei, als, ald, mbuf, sbuf, exb, Hh);
    edge_agg_kernel<<<cdiv(ehn * 32, 256), 256, 0, stream>>>(ei, exb, sbuf, hbuf, agg, Hh);
}

extern "C" void kernel_launch(void* const* d_in, const int* in_sizes, int n_in,
                              void* d_out, int out_size, void* d_ws, size_t ws_size,
                              hipStream_t stream) {
    const float* x     = (const float*)d_in[0];
    const int*   ei    = (const int*)d_in[1];   // [2, E]
    const int*   batch = (const int*)d_in[2];
    const float* W1    = (const float*)d_in[3];
    const float* a1s   = (const float*)d_in[4];
    const float* a1d   = (const float*)d_in[5];
    const float* b1    = (const float*)d_in[6];
    const float* W2    = (const float*)d_in[7];
    const float* a2s   = (const float*)d_in[8];
    const float* a2d   = (const float*)d_in[9];
    const float* b2    = (const float*)d_in[10];
    const float* W3    = (const float*)d_in[11];
    const float* a3s   = (const float*)d_in[12];
    const float* a3d   = (const float*)d_in[13];
    const float* b3    = (const float*)d_in[14];
    const float* Wc    = (const float*)d_in[15];
    const float* bc    = (const float*)d_in[16];
    float* out = (float*)d_out;

    // workspace carve (floats)
    float* ws = (float*)d_ws;
    size_t o = 0;
    float* hbuf   = ws + o; o += (size_t)NNODE * H1H * HIDC;   // reused all layers
    float* x1     = ws + o; o += (size_t)NNODE * H1H * HIDC;
    float* x2     = ws + o; o += (size_t)NNODE * H1H * HIDC;
    float* agg    = ws + o; o += (size_t)NNODE * H1H * HIDC;
    float* x3     = ws + o; o += (size_t)NNODE * HIDC;
    float* als    = ws + o; o += (size_t)NNODE * H1H;
    float* ald    = ws + o; o += (size_t)NNODE * H1H;
    float* mbuf   = ws + o; o += (size_t)NNODE * H1H;
    float* sbuf   = ws + o; o += (size_t)NNODE * H1H;
    float* exb    = ws + o; o += (size_t)ETOT * H1H;
    float* pooled = ws + o; o += (size_t)NG * HIDC;
    float* counts = ws + o; o += (size_t)NG;

    const unsigned n640 = (unsigned)NNODE * H1H * HIDC;
    const unsigned n128 = (unsigned)NNODE * HIDC;
    const dim3 gb(32, 4);                 // 4 waves, 4 M-subtiles per block

    // ---- layer 1: x [N,256] -> x1 = elu(gat(x) + b1), concat heads ----
    gemm_wmma_f32_kernel<FIN, H1H * HIDC>
        <<<dim3(cdiv(NNODE, 64), (H1H * HIDC) / 64), gb, 0, stream>>>(x, W1, hbuf, NNODE);
    run_attention(H1H, a1s, a1d, ei, hbuf, als, ald, mbuf, sbuf, exb, agg, stream);
    fin1_kernel<<<cdiv(n640, 256), 256, 0, stream>>>(agg, b1, x1, n640);

    // ---- layer 2: x2 = elu(x1 + gat(x1) + b2) ----
    gemm_wmma_f32_kernel<H1H * HIDC, H1H * HIDC>
        <<<dim3(cdiv(NNODE, 64), (H1H * HIDC) / 64), gb, 0, stream>>>(x1, W2, hbuf, NNODE);
    run_attention(H1H, a2s, a2d, ei, hbuf, als, ald, mbuf, sbuf, exb, agg, stream);
    fin2_kernel<<<cdiv(n640, 256), 256, 0, stream>>>(x1, agg, b2, x2, n640);

    // ---- layer 3: x3 = head-mean(gat(x2)) + b3 ----
    gemm_wmma_f32_kernel<H1H * HIDC, H3H * HIDC>
        <<<dim3(cdiv(NNODE, 64), (H3H * HIDC) / 64), gb, 0, stream>>>(x2, W3, hbuf, NNODE);
    run_attention(H3H, a3s, a3d, ei, hbuf, als, ald, mbuf, sbuf, exb, agg, stream);
    fin3_kernel<<<cdiv(n128, 256), 256, 0, stream>>>(agg, b3, x3, n128);

    // ---- global mean pool + classifier + log_softmax ----
    fill_kernel<<<cdiv((unsigned)NG * HIDC, 256), 256, 0, stream>>>(pooled, 0.0f, NG * HIDC);
    fill_kernel<<<1, 64, 0, stream>>>(counts, 0.0f, NG);
    pool_kernel<<<cdiv(n128, 256), 256, 0, stream>>>(x3, batch, pooled, counts);
    classify_kernel<<<1, dim3(16, NG), 0, stream>>>(pooled, counts, Wc, bc, out);
}